// ONNX_TRT_MASK_36240934043986
// MI455X (gfx1250) — compile-verified
//
#include <hip/hip_runtime.h>
#include <hip/hip_bf16.h>

// ---------------------------------------------------------------------------
// ONNX-TRT mask post-processing for MI455X (gfx1250, wave32).
// Pipeline: prep (gather/convert/top-k) -> roi_align -> WMMA upsample+softmax
//           +sigmoid mask -> gather into output tuple.
// The 14x14 -> 56x56 bilinear resize is computed as U = R * A * R^T with
// V_WMMA_F32_16X16X4_F32 (exact f32 GEMM path on the CDNA5 matrix pipes).
// ---------------------------------------------------------------------------

typedef float v2f __attribute__((ext_vector_type(2)));
typedef float v8f __attribute__((ext_vector_type(8)));

#define T_SEL   400
#define NC      80
#define NB      5
#define AR      14      // ATTN_RES
#define MR      56      // MASK_RES
#define MPIX    (MR*MR) // 3136
#define HW      160
#define NLOC    25200
#define ROW85   85
#define ROW980  980
#define MAXOBJ  100
#define BATCH   4

// -------------------------------------------------------------------- prep --
__global__ void prep_kernel(const float* __restrict__ x0,
                            const int*   __restrict__ sel,
                            float* __restrict__ selBoxes,
                            float* __restrict__ selScores,
                            float* __restrict__ selCat,
                            int*   __restrict__ idxs,
                            float* __restrict__ out /* num_det at [0..3] */) {
  __shared__ int sNumObj;
  const int tid = threadIdx.x;

  for (int t = tid; t < T_SEL; t += blockDim.x) {
    const int Xb = sel[t * 3 + 0];
    const int Yb = sel[t * 3 + 2];
    const float* row = x0 + ((size_t)Xb * NLOC + (size_t)Yb) * ROW85;
    const float b0 = row[0], b1 = row[1], b2 = row[2], b3 = row[3];
    // boxes = [cx,cy,w,h] @ CONVERT = [cx-w/2, cy-h/2, cx+w/2, cy+h/2]
    selBoxes[t * 4 + 0] = b0 - 0.5f * b2;
    selBoxes[t * 4 + 1] = b1 - 0.5f * b3;
    selBoxes[t * 4 + 2] = b0 + 0.5f * b2;
    selBoxes[t * 4 + 3] = b1 + 0.5f * b3;
    const float obj = row[4];
    float best = -3.402823e38f;
    int bi = 0;
    for (int c = 0; c < NC; ++c) {
      const float s = row[5 + c] * obj;
      if (s > best) { best = s; bi = c; }   // first-max == jnp.argmax
    }
    selScores[t] = best;
    selCat[t]    = (float)bi;
  }

  if (tid == 0) {
    // lag over T-1 entries; last = max i with lag!=0 (0 if none)
    int last = 0;
    for (int i = 0; i < T_SEL - 1; ++i) {
      const int lag = (sel[(i + 1) * 3 + 0] - sel[i * 3 + 0]) +
                      (sel[(i + 1) * 3 + 1] - sel[i * 3 + 1]) +
                      (sel[(i + 1) * 3 + 2] - sel[i * 3 + 2]);
      const int v = (lag != 0) ? i : 0;
      if (v > last) last = v;
    }
    sNumObj = last + 2;
  }
  __syncthreads();

  if (tid < BATCH) {
    const int b = tid;
    const int numObj = sNumObj;
    int cnt = 0, k = 0;
    // top_k of (t * mask) descending == collect matching t from high to low
    for (int t = T_SEL - 1; t >= 0; --t) {
      if (sel[t * 3] == b && t < numObj) {
        ++cnt;
        if (k < MAXOBJ) idxs[b * MAXOBJ + k++] = t;
      }
    }
    while (k < MAXOBJ) idxs[b * MAXOBJ + k++] = 0;
    out[b] = (float)cnt;  // num_det
  }
}

// --------------------------------------------------------------- roi_align --
__global__ void roi_align_kernel(const float* __restrict__ bases,
                                 const float* __restrict__ selBoxes,
                                 const int*   __restrict__ sel,
                                 float* __restrict__ pooled) {
  const int t = blockIdx.x / NB;
  const int c = blockIdx.x % NB;
  const int Xb = sel[t * 3];
  const float* img = bases + ((size_t)Xb * NB + c) * (HW * HW);

  const float x1 = selBoxes[t * 4 + 0] * 0.25f - 0.5f;
  const float y1 = selBoxes[t * 4 + 1] * 0.25f - 0.5f;
  const float x2 = selBoxes[t * 4 + 2] * 0.25f - 0.5f;
  const float y2 = selBoxes[t * 4 + 3] * 0.25f - 0.5f;
  const float bw = (x2 - x1) * (1.0f / MR);
  const float bh = (y2 - y1) * (1.0f / MR);

  for (int p = threadIdx.x; p < MPIX; p += blockDim.x) {
    const int oy = p / MR, ox = p % MR;
    float acc = 0.0f;
    #pragma unroll
    for (int sy = 0; sy < 2; ++sy) {
      #pragma unroll
      for (int sx = 0; sx < 2; ++sx) {
        const float gy = ((float)(oy * 2 + sy) + 0.5f) * 0.5f;
        const float gx = ((float)(ox * 2 + sx) + 0.5f) * 0.5f;
        const float yv = y1 + gy * bh;
        const float xv = x1 + gx * bw;
        const bool valid = (yv >= -1.0f) && (yv <= (float)HW) &&
                           (xv >= -1.0f) && (xv <= (float)HW);
        const float yc = fminf(fmaxf(yv, 0.0f), (float)(HW - 1));
        const float xc = fminf(fmaxf(xv, 0.0f), (float)(HW - 1));
        const int y0 = (int)floorf(yc);
        const int x0 = (int)floorf(xc);
        const int y1i = min(y0 + 1, HW - 1);
        const int x1i = min(x0 + 1, HW - 1);
        const float ly = yc - (float)y0, lx = xc - (float)x0;
        const float hy = 1.0f - ly,      hx = 1.0f - lx;
        const float v = hy * hx * img[y0  * HW + x0 ] +
                        hy * lx * img[y0  * HW + x1i] +
                        ly * hx * img[y1i * HW + x0 ] +
                        ly * lx * img[y1i * HW + x1i];
        acc += valid ? v : 0.0f;
      }
    }
    pooled[((size_t)t * NB + c) * MPIX + p] = acc * 0.25f;
  }
}

// ----------------------- WMMA upsample + softmax + pooled-mul + sigmoid ----
// One block per ROI t. 512 threads = 16 waves (wave32).
// Phase A (waves 0..4):  W[c] (16x64) = Apad[c](16x16) @ R^T(16x64)
// Phase B (all waves):   U tile (16x16) = Rtile(16x16) @ W[c] slice, c=0..4,
//                        then per-pixel softmax over c, dot with pooled,
//                        sigmoid, store mask.
__global__ void mask_wmma_kernel(const float* __restrict__ attn,
                                 const int*   __restrict__ sel,
                                 const float* __restrict__ pooled,
                                 float* __restrict__ masks) {
  __shared__ float Rlds[64][16];         // 4 KB  : R padded (rows 56..63 = 0)
  __shared__ float Apad[NB][16][16];     // 5 KB  : attn maps zero-padded
  __shared__ float Wlds[NB][16][64];     // 20 KB : intermediate A @ R^T

  const int t   = blockIdx.x;
  const int tid = threadIdx.x;
  const int Xb = sel[t * 3], Yb = sel[t * 3 + 2];
  const float* arow = attn + ((size_t)Xb * NLOC + (size_t)Yb) * ROW980;

  // Build interpolation matrix R (56x14, padded to 64x16). Edge behavior of
  // jax.image.resize (renormalized triangle kernel) == tap clamping here.
  for (int r = tid; r < 64; r += blockDim.x) {
    #pragma unroll
    for (int k = 0; k < 16; ++k) Rlds[r][k] = 0.0f;
    if (r < MR) {
      const float s  = ((float)r + 0.5f) * 0.25f - 0.5f;
      const float fj = floorf(s);
      const int   j0 = (int)fj;
      const float f  = s - fj;
      const int ja = min(max(j0, 0), AR - 1);
      const int jb = min(max(j0 + 1, 0), AR - 1);
      if (ja == jb) Rlds[r][ja] = 1.0f;
      else { Rlds[r][ja] = 1.0f - f; Rlds[r][jb] = f; }
    }
  }
  // Load 5 attention maps, zero-padded 14x14 -> 16x16
  for (int i = tid; i < NB * 256; i += blockDim.x) {
    const int c = i >> 8, rem = i & 255, y = rem >> 4, x = rem & 15;
    Apad[c][y][x] = (y < AR && x < AR) ? arow[c * (AR * AR) + y * AR + x] : 0.0f;
  }
  __syncthreads();

  const int wave = tid >> 5;
  const int lane = tid & 31;
  const int h    = lane >> 4;   // half-wave select (K split)
  const int l16  = lane & 15;

  // Phase A: W[c] = Apad[c] @ R^T   (M=16, N=64, K=16; 16 WMMAs per wave)
  if (wave < NB) {
    const int c = wave;
    v2f a[4];
    #pragma unroll
    for (int kk = 0; kk < 4; ++kk) {
      a[kk].x = Apad[c][l16][kk * 4 + 2 * h + 0];
      a[kk].y = Apad[c][l16][kk * 4 + 2 * h + 1];
    }
    #pragma unroll
    for (int nt = 0; nt < 4; ++nt) {
      v8f acc = {0.f, 0.f, 0.f, 0.f, 0.f, 0.f, 0.f, 0.f};
      #pragma unroll
      for (int kk = 0; kk < 4; ++kk) {
        v2f b;  // B[k][n] = R^T[k][n] = R[n][k]
        b.x = Rlds[nt * 16 + l16][kk * 4 + 2 * h + 0];
        b.y = Rlds[nt * 16 + l16][kk * 4 + 2 * h + 1];
        acc = __builtin_amdgcn_wmma_f32_16x16x4_f32(
            false, a[kk], false, b, (short)0, acc, false, false);
      }
      #pragma unroll
      for (int r = 0; r < 8; ++r)
        Wlds[c][r + 8 * h][nt * 16 + l16] = acc[r];
    }
  }
  __syncthreads();

  // Phase B: per wave, one 16x16 output tile for all 5 bases.
  const int Mt = wave >> 2, Nt = wave & 3;
  v2f ra[4];
  #pragma unroll
  for (int kk = 0; kk < 4; ++kk) {
    ra[kk].x = Rlds[Mt * 16 + l16][kk * 4 + 2 * h + 0];
    ra[kk].y = Rlds[Mt * 16 + l16][kk * 4 + 2 * h + 1];
  }
  v8f acc[NB];
  #pragma unroll
  for (int c = 0; c < NB; ++c) {
    acc[c] = (v8f){0.f, 0.f, 0.f, 0.f, 0.f, 0.f, 0.f, 0.f};
    #pragma unroll
    for (int kk = 0; kk < 4; ++kk) {
      v2f b;
      b.x = Wlds[c][kk * 4 + 2 * h + 0][Nt * 16 + l16];
      b.y = Wlds[c][kk * 4 + 2 * h + 1][Nt * 16 + l16];
      acc[c] = __builtin_amdgcn_wmma_f32_16x16x4_f32(
          false, ra[kk], false, b, (short)0, acc[c], false, false);
    }
  }

  const float* pBase = pooled + (size_t)t * NB * MPIX;
  #pragma unroll
  for (int r = 0; r < 8; ++r) {
    const int gm = Mt * 16 + r + 8 * h;   // C/D layout: VGPR r -> rows r, r+8
    const int gn = Nt * 16 + l16;
    if (gm < MR && gn < MR) {
      const float v0 = acc[0][r], v1 = acc[1][r], v2 = acc[2][r],
                  v3 = acc[3][r], v4 = acc[4][r];
      const float mx = fmaxf(fmaxf(fmaxf(v0, v1), fmaxf(v2, v3)), v4);
      const float e0 = expf(v0 - mx), e1 = expf(v1 - mx), e2 = expf(v2 - mx),
                  e3 = expf(v3 - mx), e4 = expf(v4 - mx);
      const float inv = 1.0f / (e0 + e1 + e2 + e3 + e4);
      const int p = gm * MR + gn;
      const float s = (e0 * pBase[0 * MPIX + p] + e1 * pBase[1 * MPIX + p] +
                       e2 * pBase[2 * MPIX + p] + e3 * pBase[3 * MPIX + p] +
                       e4 * pBase[4 * MPIX + p]) * inv;
      masks[(size_t)t * MPIX + p] = 1.0f / (1.0f + expf(-s));
    }
  }
}

// ------------------------------------------------------------------ gather --
__global__ void gather_kernel(const float* __restrict__ selBoxes,
                              const float* __restrict__ selScores,
                              const float* __restrict__ selCat,
                              const float* __restrict__ masks,
                              const int*   __restrict__ idxs,
                              float* __restrict__ out) {
  const int g   = blockIdx.x;         // b*100 + j
  const int idx = idxs[g];
  float* boxes   = out + 4;
  float* scores  = out + 4 + BATCH * MAXOBJ * 4;
  float* classes = out + 4 + BATCH * MAXOBJ * 5;
  float* dmask   = out + 4 + BATCH * MAXOBJ * 6;
  if (threadIdx.x < 4) boxes[g * 4 + threadIdx.x] = selBoxes[idx * 4 + threadIdx.x];
  if (threadIdx.x == 4) scores[g]  = selScores[idx];
  if (threadIdx.x == 5) classes[g] = selCat[idx];
  for (int p = threadIdx.x; p < MPIX; p += blockDim.x)
    dmask[(size_t)g * MPIX + p] = masks[(size_t)idx * MPIX + p];
}

// ------------------------------------------------------------------ launch --
extern "C" void kernel_launch(void* const* d_in, const int* in_sizes, int n_in,
                              void* d_out, int out_size, void* d_ws, size_t ws_size,
                              hipStream_t stream) {
  (void)in_sizes; (void)n_in; (void)out_size; (void)ws_size;
  const float* x0    = (const float*)d_in[0];
  const float* attn  = (const float*)d_in[1];
  const float* bases = (const float*)d_in[2];
  const int*   sel   = (const int*)d_in[3];
  float* out = (float*)d_out;

  float* ws        = (float*)d_ws;
  float* selBoxes  = ws;                       // 1600 f
  float* selScores = ws + 1600;                //  400 f
  float* selCat    = ws + 2000;                //  400 f
  int*   idxs      = (int*)(ws + 2400);        //  400 i
  float* pooled    = ws + 2816;                // 400*5*3136 f (~25 MB)
  float* masks     = pooled + (size_t)T_SEL * NB * MPIX;  // 400*3136 f (~5 MB)

  prep_kernel<<<1, 512, 0, stream>>>(x0, sel, selBoxes, selScores, selCat, idxs, out);
  roi_align_kernel<<<T_SEL * NB, 256, 0, stream>>>(bases, selBoxes, sel, pooled);
  mask_wmma_kernel<<<T_SEL, 512, 0, stream>>>(attn, sel, pooled, masks);
  gather_kernel<<<BATCH * MAXOBJ, 128, 0, stream>>>(selBoxes, selScores, selCat,
                                                    masks, idxs, out);
}